// AttentionLayer_42649025249822
// MI455X (gfx1250) — compile-verified
//
#include <hip/hip_runtime.h>
#include <math.h>
#include <stdint.h>

// ---------------------------------------------------------------------------
// Attention layer for MI455X (gfx1250), fp32 end-to-end via V_WMMA_F32_16X16X4_F32.
// B=4, S=16, H=16, D=128, HID=2048, P=8192 cached keys.
//
// Pipeline (all on `stream`, serialized by stream order):
//   1) gemm64<6144,scatter>  : qkv = x @ w_qkv + b_qkv  -> Q/Knew/Vnew [B,H,S,D]
//   2) attn_partial          : flash attention, split-K over key chunks.
//                              KV tiles double-buffered into LDS with
//                              GLOBAL_LOAD_ASYNC_TO_LDS_B128 (ASYNCcnt),
//                              shared by 4 batch-waves per block.
//   3) attn_combine          : merge per-chunk (O, m, l) partials
//   4) gemm64<2048,linear>   : out = attn @ w_proj + b_proj
// ---------------------------------------------------------------------------

#define BB   4
#define SS   16
#define HH   16
#define DD   128
#define HID  2048
#define PP   8192
#define NQKV 6144
#define NCHUNK 17          // 16 cached chunks (512 keys each) + 1 new-key chunk

typedef __attribute__((ext_vector_type(2))) float v2f;
typedef __attribute__((ext_vector_type(8))) float v8f;

__device__ __forceinline__ v8f wmma4(v2f a, v2f b, v8f c) {
  // D = A(16x4,f32) * B(4x16,f32) + C(16x16,f32)
  return __builtin_amdgcn_wmma_f32_16x16x4_f32(false, a, false, b, (short)0, c,
                                               false, false);
}

// ---------------------------------------------------------------------------
// GEMM: out[64, NCOLS] = A[64, 2048] @ W[2048, NCOLS] + bias
// One wave per block; block n covers columns [16n, 16n+16), all 64 rows
// (4 accumulator tiles). SCATTER=true splits columns into Q/K/V [B,H,S,D].
// ---------------------------------------------------------------------------
template <int NCOLS, bool SCATTER>
__global__ __launch_bounds__(32) void gemm64_kernel(
    const float* __restrict__ A, const float* __restrict__ W,
    const float* __restrict__ bias, float* __restrict__ out0,
    float* __restrict__ out1, float* __restrict__ out2) {
  const int lane = threadIdx.x & 31;
  const int half = lane >> 4;   // which 16-lane half
  const int lm   = lane & 15;
  const int n0   = blockIdx.x * 16;

  v8f acc[4] = {};

  for (int k0 = 0; k0 < HID; k0 += 4) {
    const int ka = k0 + 2 * half;          // this lane's K pair
    v2f bf;
    bf.x = W[(size_t)(ka + 0) * NCOLS + n0 + lm];
    bf.y = W[(size_t)(ka + 1) * NCOLS + n0 + lm];
#pragma unroll
    for (int mt = 0; mt < 4; ++mt) {
      v2f af = *(const v2f*)(A + (size_t)(mt * 16 + lm) * HID + ka);
      acc[mt] = wmma4(af, bf, acc[mt]);
    }
  }

  const float bv = bias[n0 + lm];
  if (SCATTER) {
    const int sel   = n0 / HID;        // 0=q 1=k 2=v (uniform per block)
    const int hid_i = n0 % HID;
    const int hh    = hid_i >> 7;      // head (uniform)
    const int dd0   = hid_i & 127;     // d base
    float* dst = (sel == 0) ? out0 : ((sel == 1) ? out1 : out2);
#pragma unroll
    for (int mt = 0; mt < 4; ++mt) {
#pragma unroll
      for (int r = 0; r < 8; ++r) {
        const int m  = mt * 16 + r + 8 * half;   // global row = b*16+s
        const int bb = m >> 4, sgl = m & 15;
        dst[((((size_t)bb * HH) + hh) * SS + sgl) * DD + dd0 + lm] =
            acc[mt][r] + bv;
      }
    }
  } else {
#pragma unroll
    for (int mt = 0; mt < 4; ++mt) {
#pragma unroll
      for (int r = 0; r < 8; ++r) {
        const int m = mt * 16 + r + 8 * half;
        out0[(size_t)m * NCOLS + n0 + lm] = acc[mt][r] + bv;
      }
    }
  }
}

// ---------------------------------------------------------------------------
// Async stage of one 16x128 K tile and one 16x128 V tile into LDS.
// Each of the 128 threads copies 16 floats per tensor as 4x b128 async ops
// (8 wave-instructions per tile per wave -> ASYNCcnt += 8).
// ---------------------------------------------------------------------------
__device__ __forceinline__ void stage_tile_async(const float* __restrict__ ck,
                                                 const float* __restrict__ cv,
                                                 float* ldsKbuf, float* ldsVbuf,
                                                 int kb, int h, int tid) {
  const int i  = tid >> 3;          // key row 0..15
  const int d0 = (tid & 7) * 16;    // 16 floats per thread
  const float* sk = ck + (((size_t)(kb + i)) * HH + h) * DD + d0;
  const float* sv = cv + (((size_t)(kb + i)) * HH + h) * DD + d0;
  // Low 32 bits of a generic LDS pointer == LDS byte address (aperture rule).
  const uint32_t dk = (uint32_t)(uintptr_t)(ldsKbuf + i * 128 + d0);
  const uint32_t dv = (uint32_t)(uintptr_t)(ldsVbuf + i * 128 + d0);
#pragma unroll
  for (int q4 = 0; q4 < 4; ++q4) {
    asm volatile("global_load_async_to_lds_b128 %0, %1, off"
                 :
                 : "v"(dk + q4 * 16), "v"(sk + q4 * 4)
                 : "memory");
    asm volatile("global_load_async_to_lds_b128 %0, %1, off"
                 :
                 : "v"(dv + q4 * 16), "v"(sv + q4 * 4)
                 : "memory");
  }
}

// ---------------------------------------------------------------------------
// Flash attention partial. grid = H * NCHUNK blocks, 128 threads (4 waves).
// Wave w handles batch b=w; cached K/V tiles are async-copied to LDS
// (double-buffered) and shared by all 4 batches -> KV cache read 1x from HBM.
// Chunk 16 handles the S=16 new keys (per-batch, loaded directly).
// Writes un-normalized O partials plus per-row (m, l) stats.
// ---------------------------------------------------------------------------
__global__ __launch_bounds__(128) void attn_partial_kernel(
    const float* __restrict__ Q, const float* __restrict__ Kn,
    const float* __restrict__ Vn, const float* __restrict__ ck,
    const float* __restrict__ cv, float* __restrict__ Opart,
    float* __restrict__ Mpart, float* __restrict__ Lpart) {
  __shared__ alignas(16) float ldsK[2][16 * 128];
  __shared__ alignas(16) float ldsV[2][16 * 128];
  __shared__ alignas(16) float pt[4][16 * 16];

  const int tid  = threadIdx.x;
  const int wave = tid >> 5;   // = batch b
  const int lane = tid & 31;
  const int half = lane >> 4;
  const int lm   = lane & 15;

  const int c = blockIdx.x % NCHUNK;
  const int h = blockIdx.x / NCHUNK;
  const int b = wave;

  const float scale = 0.08838834764831845f;  // 1/sqrt(128)

  // Preload Q fragments (scale folded in). A-frag: lane=query row, half picks K pair.
  v2f qf[32];
  const float* qrow = Q + (((size_t)b * HH + h) * SS + lm) * DD;
#pragma unroll
  for (int kk = 0; kk < 32; ++kk) {
    qf[kk] = *(const v2f*)(qrow + kk * 4 + 2 * half);
    qf[kk] *= scale;
  }

  v8f o[8] = {};
  float rm[8], rl[8];
#pragma unroll
  for (int r = 0; r < 8; ++r) { rm[r] = -3.0e38f; rl[r] = 0.0f; }

  if (c < 16) {
    const int kstart = c * 512;

    // Prologue: stage first tile into buffer 0.
    stage_tile_async(ck, cv, ldsK[0], ldsV[0], kstart, h, tid);

    for (int it = 0; it < 32; ++it) {
      const int kb  = kstart + it * 16;
      const int cur = it & 1;
      // All waves have finished reading buffer cur^1 (iteration it-1), so it
      // is safe to start overwriting it with tile it+1.
      __syncthreads();
      if (it + 1 < 32) {
        stage_tile_async(ck, cv, ldsK[cur ^ 1], ldsV[cur ^ 1], kb + 16, h, tid);
        // Async "done" returns in order: <=8 outstanding means tile `it`
        // (issued 8 ops ago) has fully landed in LDS; tile it+1 stays in flight.
        asm volatile("s_wait_asynccnt 0x8" ::: "memory");
      } else {
        asm volatile("s_wait_asynccnt 0x0" ::: "memory");
      }
      __syncthreads();   // current buffer valid for every wave

      const float* tK = ldsK[cur];
      const float* tV = ldsV[cur];

      // S tile = Q @ K^T  (B-frag: lane=key, half picks d pair)
      v8f s = {};
#pragma unroll
      for (int kk = 0; kk < 32; ++kk) {
        v2f bf = *(const v2f*)&tK[lm * 128 + kk * 4 + 2 * half];
        s = wmma4(qf[kk], bf, s);
      }

      // online softmax (rows r+8*half; 16-lane reductions stay within halves)
      float al[8];
#pragma unroll
      for (int r = 0; r < 8; ++r) {
        float sv2 = s[r];
        float mx = sv2;
        mx = fmaxf(mx, __shfl_xor(mx, 1));
        mx = fmaxf(mx, __shfl_xor(mx, 2));
        mx = fmaxf(mx, __shfl_xor(mx, 4));
        mx = fmaxf(mx, __shfl_xor(mx, 8));
        const float mn = fmaxf(rm[r], mx);
        const float p  = __expf(sv2 - mn);
        float ps = p;
        ps += __shfl_xor(ps, 1);
        ps += __shfl_xor(ps, 2);
        ps += __shfl_xor(ps, 4);
        ps += __shfl_xor(ps, 8);
        al[r] = __expf(rm[r] - mn);
        rl[r] = rl[r] * al[r] + ps;
        rm[r] = mn;
        pt[wave][(r + 8 * half) * 16 + lm] = p;  // transpose P via LDS
      }
#pragma unroll
      for (int t = 0; t < 8; ++t)
#pragma unroll
        for (int r = 0; r < 8; ++r) o[t][r] *= al[r];

      // O += P @ V
      v2f pf[4];
#pragma unroll
      for (int j = 0; j < 4; ++j)
        pf[j] = *(const v2f*)&pt[wave][lm * 16 + j * 4 + 2 * half];
#pragma unroll
      for (int t = 0; t < 8; ++t) {
#pragma unroll
        for (int j = 0; j < 4; ++j) {
          const int kr = j * 4 + 2 * half;
          v2f bf;
          bf.x = tV[kr * 128 + t * 16 + lm];
          bf.y = tV[(kr + 1) * 128 + t * 16 + lm];
          o[t] = wmma4(pf[j], bf, o[t]);
        }
      }
    }
  } else {
    // Chunk 16: the 16 new (per-batch) keys, straight from global.
    const float* krow = Kn + (((size_t)b * HH + h) * SS + lm) * DD;
    v8f s = {};
#pragma unroll
    for (int kk = 0; kk < 32; ++kk) {
      v2f bf = *(const v2f*)(krow + kk * 4 + 2 * half);
      s = wmma4(qf[kk], bf, s);
    }
    float al[8];
#pragma unroll
    for (int r = 0; r < 8; ++r) {
      float sv2 = s[r];
      float mx = sv2;
      mx = fmaxf(mx, __shfl_xor(mx, 1));
      mx = fmaxf(mx, __shfl_xor(mx, 2));
      mx = fmaxf(mx, __shfl_xor(mx, 4));
      mx = fmaxf(mx, __shfl_xor(mx, 8));
      const float mn = fmaxf(rm[r], mx);
      const float p  = __expf(sv2 - mn);
      float ps = p;
      ps += __shfl_xor(ps, 1);
      ps += __shfl_xor(ps, 2);
      ps += __shfl_xor(ps, 4);
      ps += __shfl_xor(ps, 8);
      al[r] = __expf(rm[r] - mn);
      rl[r] = rl[r] * al[r] + ps;
      rm[r] = mn;
      pt[wave][(r + 8 * half) * 16 + lm] = p;
    }
#pragma unroll
    for (int t = 0; t < 8; ++t)
#pragma unroll
      for (int r = 0; r < 8; ++r) o[t][r] *= al[r];

    v2f pf[4];
#pragma unroll
    for (int j = 0; j < 4; ++j)
      pf[j] = *(const v2f*)&pt[wave][lm * 16 + j * 4 + 2 * half];
#pragma unroll
    for (int t = 0; t < 8; ++t) {
#pragma unroll
      for (int j = 0; j < 4; ++j) {
        const int kr = j * 4 + 2 * half;
        const float* vr = Vn + (((size_t)b * HH + h) * SS + kr) * DD;
        v2f bf;
        bf.x = vr[t * 16 + lm];
        bf.y = vr[DD + t * 16 + lm];   // next key row
        o[t] = wmma4(pf[j], bf, o[t]);
      }
    }
  }

  // Store un-normalized partials + stats for this (b,h,chunk).
  const int bh = b * HH + h;
  float* op = Opart + ((size_t)bh * NCHUNK + c) * (SS * DD);
#pragma unroll
  for (int t = 0; t < 8; ++t)
#pragma unroll
    for (int r = 0; r < 8; ++r)
      op[(r + 8 * half) * DD + t * 16 + lm] = o[t][r];
  if (lm == 0) {
    float* mp = Mpart + ((size_t)bh * NCHUNK + c) * SS + 8 * half;
    float* lp = Lpart + ((size_t)bh * NCHUNK + c) * SS + 8 * half;
#pragma unroll
    for (int r = 0; r < 8; ++r) { mp[r] = rm[r]; lp[r] = rl[r]; }
  }
}

// ---------------------------------------------------------------------------
// Combine split-K partials: one block per (b,h); 256 threads = 16 rows x 16
// d-groups of 8. Writes attn output as [B,S,HID] for the final projection.
// ---------------------------------------------------------------------------
__global__ __launch_bounds__(256) void attn_combine_kernel(
    const float* __restrict__ Opart, const float* __restrict__ Mpart,
    const float* __restrict__ Lpart, float* __restrict__ Aout) {
  const int bh = blockIdx.x;
  const int h = bh & 15, b = bh >> 4;
  const int tid = threadIdx.x;
  const int m  = tid >> 4;
  const int d0 = (tid & 15) * 8;

  float gmax = -3.0e38f;
  for (int cc = 0; cc < NCHUNK; ++cc)
    gmax = fmaxf(gmax, Mpart[((size_t)bh * NCHUNK + cc) * SS + m]);

  float den = 0.0f;
  float acc[8];
#pragma unroll
  for (int i = 0; i < 8; ++i) acc[i] = 0.0f;

  for (int cc = 0; cc < NCHUNK; ++cc) {
    const float wgt = __expf(Mpart[((size_t)bh * NCHUNK + cc) * SS + m] - gmax);
    den += Lpart[((size_t)bh * NCHUNK + cc) * SS + m] * wgt;
    const float* op =
        Opart + ((size_t)bh * NCHUNK + cc) * (SS * DD) + m * DD + d0;
#pragma unroll
    for (int i = 0; i < 8; ++i) acc[i] += op[i] * wgt;
  }
  const float inv = 1.0f / den;
  float* dst = Aout + ((size_t)(b * SS + m)) * HID + h * DD + d0;
#pragma unroll
  for (int i = 0; i < 8; ++i) dst[i] = acc[i] * inv;
}

// ---------------------------------------------------------------------------
extern "C" void kernel_launch(void* const* d_in, const int* in_sizes, int n_in,
                              void* d_out, int out_size, void* d_ws,
                              size_t ws_size, hipStream_t stream) {
  const float* x     = (const float*)d_in[0];
  const float* ck    = (const float*)d_in[1];
  const float* cv    = (const float*)d_in[2];
  const float* wqkv  = (const float*)d_in[3];
  const float* bqkv  = (const float*)d_in[4];
  const float* wproj = (const float*)d_in[5];
  const float* bproj = (const float*)d_in[6];
  float* out = (float*)d_out;

  float* ws   = (float*)d_ws;
  float* Qw   = ws;                                     // [B,H,S,D]
  float* Kn   = Qw + (size_t)BB * HH * SS * DD;         // [B,H,S,D]
  float* Vn   = Kn + (size_t)BB * HH * SS * DD;         // [B,H,S,D]
  float* Opart = Vn + (size_t)BB * HH * SS * DD;        // [BH,NCHUNK,S,D]
  float* Mpart = Opart + (size_t)BB * HH * NCHUNK * SS * DD;
  float* Lpart = Mpart + (size_t)BB * HH * NCHUNK * SS;
  float* Aout  = Lpart + (size_t)BB * HH * NCHUNK * SS; // [B,S,HID]

  // 1) fused QKV projection (scatter into Q/Knew/Vnew)
  gemm64_kernel<NQKV, true><<<NQKV / 16, 32, 0, stream>>>(x, wqkv, bqkv, Qw, Kn,
                                                          Vn);
  // 2) flash attention partials (split-K, async double-buffered KV in LDS)
  attn_partial_kernel<<<HH * NCHUNK, 128, 0, stream>>>(Qw, Kn, Vn, ck, cv,
                                                       Opart, Mpart, Lpart);
  // 3) combine partials
  attn_combine_kernel<<<BB * HH, 256, 0, stream>>>(Opart, Mpart, Lpart, Aout);
  // 4) output projection
  gemm64_kernel<HID, false><<<HID / 16, 32, 0, stream>>>(Aout, wproj, bproj,
                                                         out, nullptr, nullptr);
}